// Ghost_vlad_pooling_68959994904734
// MI455X (gfx1250) — compile-verified
//
#include <hip/hip_runtime.h>
#include <math.h>

typedef __attribute__((ext_vector_type(2))) float v2f;
typedef __attribute__((ext_vector_type(8))) float v8f;

#define KG     10          // K_CENTERS + G_CENTERS
#define KC     8           // K_CENTERS (output rows)
#define HW     1024        // 16*64 pixels per batch
#define DDIM   512
#define BATCH  32
#define NT     (HW / 4)    // 256 hw-chunks of 4 (WMMA K=4)
#define DTILES (DDIM / 16) // 32 d-tiles of 16

// ---------------------------------------------------------------------------
// Kernel 1: per-pixel softmax over KG=10 scores, padded to 16 rows with zeros,
// written in the V_WMMA_F32_16X16X4_F32 A-operand swizzle:
//   A tile (M=16 x K=4), lane l<16 holds {K=0,K=1} for M=l,
//                        lane l>=16 holds {K=2,K=3} for M=l-16.
// Layout in ws: As[b][t][lane][slot], slot = hw%2 within the K-half.
// ---------------------------------------------------------------------------
__global__ __launch_bounds__(256) void ghostvlad_softmax_swizzle(
    const float* __restrict__ score, float* __restrict__ As)
{
  int gid = blockIdx.x * 256 + threadIdx.x;   // 0 .. B*HW-1
  int b = gid >> 10;
  int p = gid & 1023;
  int t = p >> 2;
  int r = p & 3;

  const float* sc = score + (size_t)gid * KG;
  float m = sc[0];
  #pragma unroll
  for (int i = 1; i < KG; ++i) m = fmaxf(m, sc[i]);
  float e[KG];
  float s = 0.0f;
  #pragma unroll
  for (int i = 0; i < KG; ++i) { e[i] = expf(sc[i] - m); s += e[i]; }
  float inv = 1.0f / s;

  float* dst = As + (size_t)(b * NT + t) * 64;   // 32 lanes * 2 slots
  int lane_base = (r >= 2) ? 16 : 0;
  int slot = r & 1;
  #pragma unroll
  for (int k = 0; k < 16; ++k) {
    float v = (k < KG) ? e[k] * inv : 0.0f;
    dst[(k + lane_base) * 2 + slot] = v;
  }
}

// ---------------------------------------------------------------------------
// Kernel 2: one wave per (batch, d-tile).  C(16x16) += A(16x4) x B(4x16) over
// 256 hw-chunks using v_wmma_f32_16x16x4_f32.  a_sum[k] computed on the fly
// from the A tiles (xor-16 lane combine).  Rows k=0..7 (lanes 0..15 of the
// C layout) get the a_sum*cluster subtraction and are stored to d_out.
// ---------------------------------------------------------------------------
__global__ __launch_bounds__(256) void ghostvlad_wmma_gemm(
    const float* __restrict__ feat, const float* __restrict__ As,
    const float* __restrict__ cluster, float* __restrict__ out)
{
  int wave = (blockIdx.x * 256 + threadIdx.x) >> 5;  // 0..1023
  int lane = threadIdx.x & 31;
  int b    = wave >> 5;          // 32 d-tiles per batch
  int d0   = (wave & 31) << 4;   // d-tile start column
  int ln   = lane & 15;          // N / M index within half
  int roff = (lane >> 4) << 1;   // K rows {0,1} vs {2,3}

  const float* ap = As + (size_t)(b * NT) * 64 + lane * 2;
  const float* fp = feat + ((size_t)b * HW + roff) * DDIM + d0 + ln;

  v8f acc = {};
  float asum = 0.0f;

  #pragma unroll 4
  for (int t = 0; t < NT; ++t) {
    v2f a = *(const v2f*)ap;                 // 8B swizzled A load
    v2f bb;
    bb.x = fp[0];                            // feat[4t+roff  ][d0+ln]
    bb.y = fp[DDIM];                         // feat[4t+roff+1][d0+ln]
    __builtin_prefetch(fp + 8 * 4 * DDIM, 0, 0);   // global_prefetch_b8
    acc = __builtin_amdgcn_wmma_f32_16x16x4_f32(
        false, a, false, bb, (short)0, acc, false, false);
    asum += a.x + a.y;                       // partial a_sum for this K-half
    ap += 64;
    fp += 4 * DDIM;
  }

  // combine K-halves: lane k (k<16) now holds a_sum[k]
  float asum_full = asum + __shfl_xor(asum, 16, 32);

  // C/D layout: VGPR v, lanes 0-15 -> row M=v, col N=lane.
  // Output only needs rows 0..7 -> entirely in lanes 0..15.
  if (lane < 16) {
    float* orow = out + (size_t)b * KC * DDIM + d0 + lane;
    #pragma unroll
    for (int v = 0; v < KC; ++v) {
      float av = __shfl(asum_full, v, 32);
      float cl = cluster[v * DDIM + d0 + lane];
      orow[v * DDIM] = acc[v] - av * cl;
    }
  }
}

// ---------------------------------------------------------------------------
// Kernel 3: in-place L2 normalization of each (b,k) row of 512 floats.
// Fixed-order LDS tree reduction -> deterministic.
// ---------------------------------------------------------------------------
__global__ __launch_bounds__(128) void ghostvlad_l2norm(float* __restrict__ out)
{
  __shared__ float red[128];
  float* p = out + (size_t)blockIdx.x * DDIM;
  int tid = threadIdx.x;

  float v0 = p[tid];
  float v1 = p[tid + 128];
  float v2 = p[tid + 256];
  float v3 = p[tid + 384];
  red[tid] = v0 * v0 + v1 * v1 + v2 * v2 + v3 * v3;
  __syncthreads();
  #pragma unroll
  for (int off = 64; off > 0; off >>= 1) {
    if (tid < off) red[tid] += red[tid + off];
    __syncthreads();
  }
  float inv = 1.0f / sqrtf(fmaxf(red[0], 1e-12f));
  p[tid]       = v0 * inv;
  p[tid + 128] = v1 * inv;
  p[tid + 256] = v2 * inv;
  p[tid + 384] = v3 * inv;
}

// ---------------------------------------------------------------------------
extern "C" void kernel_launch(void* const* d_in, const int* in_sizes, int n_in,
                              void* d_out, int out_size, void* d_ws, size_t ws_size,
                              hipStream_t stream) {
  const float* feat    = (const float*)d_in[0];  // (32,16,64,512) f32
  const float* score   = (const float*)d_in[1];  // (32,16,64,10)  f32
  const float* cluster = (const float*)d_in[2];  // (10,512)       f32
  float* out = (float*)d_out;                    // (32, 8*512)    f32
  float* As  = (float*)d_ws;                     // 32*256*64 floats = 2 MB

  // 1) softmax -> WMMA-swizzled padded A (fully overwrites its ws region)
  ghostvlad_softmax_swizzle<<<(BATCH * HW) / 256, 256, 0, stream>>>(score, As);

  // 2) 1024 waves (128 blocks x 8 waves): WMMA GEMM + a_sum*cluster subtract
  ghostvlad_wmma_gemm<<<(BATCH * DTILES * 32) / 256, 256, 0, stream>>>(
      feat, As, cluster, out);

  // 3) one block per (b,k) row: in-place L2 normalize
  ghostvlad_l2norm<<<BATCH * KC, 128, 0, stream>>>(out);
}